// System2Reasoner_67654324846913
// MI455X (gfx1250) — compile-verified
//
#include <hip/hip_runtime.h>
#include <hip/hip_bf16.h>

typedef __attribute__((ext_vector_type(2))) float v2f;
typedef __attribute__((ext_vector_type(8))) float v8f;

#define P_TOT 2048
#define M_TOT 100000
#define DIM   128
#define TOPK  50
#define ROWS  16                    // test rows per workgroup (one WMMA M-tile)
#define WAVES 8                     // waves per workgroup (256 threads, wave32)
#define TILES (M_TOT / 16)          // 6250 exact 16-node tiles
#define PAIRS (TILES / 2)           // 3125 pairs of tiles (32 nodes each)

// ---------------------------------------------------------------------------
// Kernel 1: sim = X @ Mem^T via V_WMMA_F32_16X16X4_F32, streaming per-row
// top-50 selection (never materializes the 2048x100000 sim matrix).
//
// Each wave processes TWO 16-node tiles per iteration with even/odd K-step
// splitting -> 4 independent WMMA accumulator chains (c0,d0,c1,d1) so the
// matrix pipe is never stalled on a single RAW chain or a single B fetch.
//
// WMMA f32 16x16x4 register layouts (ISA 7.12.2):
//   A (16x4):  lanes 0-15 -> rows M=0..15; lane l holds (K=4s,4s+1) for l<16
//              and (K=4s+2,4s+3) for l>=16 -> one b64 per K-step.
//   B (4x16):  same striping with N across lanes -> per-lane b64 from the
//              memory-node row (lanes l and l+16 read adjacent 8B pairs).
//   C/D (16x16 f32, 8 VGPRs): VGPR j, lanes 0-15 -> (M=j, N=lane),
//              lanes 16-31 -> (M=j+8, N=lane-16).
// ---------------------------------------------------------------------------
__global__ __launch_bounds__(256)
void simtopk_kernel(const float* __restrict__ X, const float* __restrict__ Mem,
                    int* __restrict__ eidx_mem, int* __restrict__ eidx_test)
{
    __shared__ float topv[WAVES][ROWS][TOPK];   // per-wave per-row top values
    __shared__ int   topi[WAVES][ROWS][TOPK];   // matching memory indices
    __shared__ float minv[WAVES][ROWS];         // current min of each 50-list
    __shared__ int   minp[WAVES][ROWS];         // its position

    const int tid  = threadIdx.x;
    const int wv   = tid >> 5;
    const int lane = tid & 31;
    const int half = lane >> 4;     // 0: lanes 0-15, 1: lanes 16-31
    const int col  = lane & 15;
    const int r0   = blockIdx.x * ROWS;

    for (int i = tid; i < WAVES * ROWS * TOPK; i += blockDim.x) {
        ((float*)topv)[i] = -__builtin_inff();
        ((int*)topi)[i]   = 0;
    }
    if (tid < WAVES * ROWS) {
        ((float*)minv)[tid] = -__builtin_inff();
        ((int*)minp)[tid]   = 0;
    }
    __syncthreads();

    // Preload the A panel (16 rows x 128 K) into 64 VGPRs in WMMA layout.
    const float* ap = X + (size_t)(r0 + col) * DIM + (half << 1);
    v2f a[32];
#pragma unroll
    for (int s = 0; s < 32; ++s) a[s] = *(const v2f*)(ap + 4 * s);

    const float* bbase = Mem + (half << 1);
    const unsigned hmask = half ? 0xFFFF0000u : 0x0000FFFFu;

    for (int tp = wv; tp < PAIRS; tp += WAVES) {
        const int nbase = tp * 32;
        const float* bp0 = bbase + (size_t)(nbase + col) * DIM;
        const float* bp1 = bp0 + 16 * DIM;

        v8f c0 = {0.f,0.f,0.f,0.f,0.f,0.f,0.f,0.f};
        v8f d0 = c0, c1 = c0, d1 = c0;
#pragma unroll
        for (int s = 0; s < 32; s += 2) {
            v2f b00 = *(const v2f*)(bp0 + 4 * s);
            v2f b10 = *(const v2f*)(bp1 + 4 * s);
            v2f b01 = *(const v2f*)(bp0 + 4 * (s + 1));
            v2f b11 = *(const v2f*)(bp1 + 4 * (s + 1));
            c0 = __builtin_amdgcn_wmma_f32_16x16x4_f32(false, a[s],     false, b00, (short)0, c0, false, false);
            c1 = __builtin_amdgcn_wmma_f32_16x16x4_f32(false, a[s],     false, b10, (short)0, c1, false, false);
            d0 = __builtin_amdgcn_wmma_f32_16x16x4_f32(false, a[s + 1], false, b01, (short)0, d0, false, false);
            d1 = __builtin_amdgcn_wmma_f32_16x16x4_f32(false, a[s + 1], false, b11, (short)0, d1, false, false);
        }
        v8f cc0, cc1;
#pragma unroll
        for (int j = 0; j < 8; ++j) { cc0[j] = c0[j] + d0[j]; cc1[j] = c1[j] + d1[j]; }

        // Streaming top-50 insert for both tiles. Each half-wave owns rows
        // j (half 0) / j+8 (half 1); a row's 16 candidates sit in one half.
#pragma unroll
        for (int tt = 0; tt < 2; ++tt) {
            const int nb = nbase + tt * 16;
#pragma unroll
            for (int j = 0; j < 8; ++j) {
                const int row = j + (half << 3);
                float v = tt ? cc1[j] : cc0[j];
                float tmin = minv[wv][row];
                unsigned mask = (unsigned)__ballot(v > tmin) & hmask;
                while (mask) {                       // mask uniform within a half
                    int src = __ffs(mask) - 1;
                    mask &= mask - 1;
                    float vv = __shfl(v, src);       // broadcast candidate
                    int   ii = nb + (src & 15);
                    float cur = minv[wv][row];
                    if (vv > cur) {                  // uniform within the half
                        int mp = minp[wv][row];
                        if (col == 0) { topv[wv][row][mp] = vv; topi[wv][row][mp] = ii; }
                        // Cooperative rescan for the new min; entry mp is
                        // substituted from registers (no LDS RAW dependence).
                        float m = __builtin_inff(); int p = 64;
                        for (int e = col; e < TOPK; e += 16) {
                            float x = (e == mp) ? vv : topv[wv][row][e];
                            if (x < m) { m = x; p = e; }
                        }
                        for (int off = 8; off > 0; off >>= 1) {
                            float om = __shfl_xor(m, off, 16);
                            int   op = __shfl_xor(p, off, 16);
                            if (om < m || (om == m && op < p)) { m = om; p = op; }
                        }
                        if (col == 0) { minv[wv][row] = m; minp[wv][row] = p; }
                    }
                }
            }
        }
    }
    __syncthreads();

    // Merge the 8 per-wave lists (400 candidates) per row -> sorted top-50.
    // Wave wv owns rows 2*wv and 2*wv+1.
    for (int rr = 0; rr < 2; ++rr) {
        const int row   = wv * 2 + rr;
        const int pglob = r0 + row;
        for (int k = 0; k < TOPK; ++k) {
            float bm = -__builtin_inff(); int bq = 0x7FFFFFFF;
            for (int q = lane; q < WAVES * TOPK; q += 32) {
                float x = topv[q / TOPK][row][q % TOPK];
                if (x > bm || (x == bm && q < bq)) { bm = x; bq = q; }
            }
            for (int off = 16; off > 0; off >>= 1) {
                float om = __shfl_xor(bm, off);
                int   oq = __shfl_xor(bq, off);
                if (om > bm || (om == bm && oq < bq)) { bm = om; bq = oq; }
            }
            int widx = topi[bq / TOPK][row][bq % TOPK];   // bq uniform now
            if (lane == 0) {
                eidx_mem[pglob * TOPK + k]  = widx;       // edge_index row 0
                eidx_test[pglob * TOPK + k] = pglob;      // edge_index row 1
                topv[bq / TOPK][row][bq % TOPK] = -__builtin_inff();
            }
        }
    }
}

// ---------------------------------------------------------------------------
// Kernel 2: token evidence pooling. 34 MFLOP total -> single workgroup.
// logits = relu(X@W1+b1)@W2+b2 ; softmax over P ; g = sum_p w_p X_p ; L2-norm.
// ---------------------------------------------------------------------------
__global__ __launch_bounds__(256)
void pool_kernel(const float* __restrict__ X,  const float* __restrict__ W1,
                 const float* __restrict__ b1, const float* __restrict__ W2,
                 const float* __restrict__ b2, float* __restrict__ gout)
{
    __shared__ float lg[P_TOT];
    __shared__ float red[256];
    __shared__ float gacc[DIM];
    const int tid = threadIdx.x;

    for (int p = tid; p < P_TOT; p += 256) {
        const float* xp = X + (size_t)p * DIM;
        float l = b2[0];
        for (int h = 0; h < DIM / 2; ++h) {
            float s = b1[h];
#pragma unroll 8
            for (int k = 0; k < DIM; ++k) s = fmaf(xp[k], W1[k * (DIM / 2) + h], s);
            l = fmaf(fmaxf(s, 0.f), W2[h], l);
        }
        lg[p] = l;
    }
    __syncthreads();

    float m = -__builtin_inff();
    for (int p = tid; p < P_TOT; p += 256) m = fmaxf(m, lg[p]);
    red[tid] = m; __syncthreads();
    for (int o = 128; o > 0; o >>= 1) {
        if (tid < o) red[tid] = fmaxf(red[tid], red[tid + o]);
        __syncthreads();
    }
    const float gm = red[0]; __syncthreads();

    float s = 0.f;
    for (int p = tid; p < P_TOT; p += 256) s += expf(lg[p] - gm);
    red[tid] = s; __syncthreads();
    for (int o = 128; o > 0; o >>= 1) {
        if (tid < o) red[tid] += red[tid + o];
        __syncthreads();
    }
    const float Z = red[0]; __syncthreads();

    for (int p = tid; p < P_TOT; p += 256) lg[p] = expf(lg[p] - gm) / Z;
    __syncthreads();

    if (tid < DIM) {
        float acc = 0.f;
        for (int p = 0; p < P_TOT; ++p)
            acc = fmaf(lg[p], X[(size_t)p * DIM + tid], acc);
        gacc[tid] = acc;
        red[tid] = acc * acc;
    }
    __syncthreads();
    for (int o = 64; o > 0; o >>= 1) {
        if (tid < o) red[tid] += red[tid + o];
        __syncthreads();
    }
    const float nrm = sqrtf(red[0]);
    const float scale = 1.f / fmaxf(nrm, 1e-12f);
    if (tid < DIM) gout[tid] = gacc[tid] * scale;
}

// ---------------------------------------------------------------------------
extern "C" void kernel_launch(void* const* d_in, const int* in_sizes, int n_in,
                              void* d_out, int out_size, void* d_ws, size_t ws_size,
                              hipStream_t stream)
{
    (void)in_sizes; (void)n_in; (void)out_size; (void)d_ws; (void)ws_size;
    const float* X   = (const float*)d_in[0];   // [2048,128]
    const float* Mem = (const float*)d_in[1];   // [100000,128]
    const float* W1  = (const float*)d_in[2];   // [128,64]
    const float* b1  = (const float*)d_in[3];   // [64]
    const float* W2  = (const float*)d_in[4];   // [64,1]
    const float* b2  = (const float*)d_in[5];   // [1]

    // d_out layout: edge_index int32 [2, P*K] then g float32 [128]
    int*   e0   = (int*)d_out;                      // memory indices
    int*   e1   = e0 + P_TOT * TOPK;                // test indices
    float* gout = (float*)d_out + 2 * P_TOT * TOPK; // pooled vector

    simtopk_kernel<<<P_TOT / ROWS, 256, 0, stream>>>(X, Mem, e0, e1);
    pool_kernel<<<1, 256, 0, stream>>>(X, W1, b1, W2, b2, gout);
}